// SE3TransformerWrapper_20770461843842
// MI455X (gfx1250) — compile-verified
//
#include <hip/hip_runtime.h>
#include <hip/hip_bf16.h>
#include <math.h>

// ============================================================================
// SE(3)-Transformer forward for MI455X (gfx1250, wave32).
//
//  * All f0[src]@W edge matmuls hoisted to per-node GEMMs (32x FLOP cut),
//    gathered per edge. Degree-1 paths factored through per-node T0/U/SelfU.
//  * ef@Wek and ef@Wev computed in ONE pass over edge_feat (dual-B WMMA GEMM)
//    -> edge_feat (128 MB) streamed once per layer.
//  * B fragments preloaded to registers; K-loop = 1 b64 A load + WMMAs.
//  * Segment softmax: 3 edge passes with L2-resident atomics
//    (monotone float->uint map + atomic_max_u32; atomic_add_f32 elsewhere).
//
// Input flattening assumption (setup_inputs insertion order, params recursed):
//  0:f0[N,32] 1:coords[N,3] 2:edge_feat[E,32]
//  3..9  l1: Wq Wk Wek Wv0 Wev W01 Wself0          (all [32,32])
//  10..19 l2: Wq Wk Wek Wv0 Wev W01[32,8] Wself0 W10 W11[32,8] Wself1[32,8]
//  20:Wwo[32,8] 21:bwo[8] 22:Wwb[32,8] 23:Wwc1[32,32] 24:bwc1[32]
//  25:Wwc2[32,15] 26:Wab[32,1] 27:bab[1] 28:Wc[15,32,2]
//  n_in-2: edge_src[E] (int)   n_in-1: edge_dst[E] (int)
// Output (concat): wo[N,8] wb[N,8] wc[N,15] ampl[N,1] cs[15,N,2] h0[N,32] h1[N,8,3]
// ============================================================================

typedef __attribute__((ext_vector_type(2))) float v2f;
typedef __attribute__((ext_vector_type(8))) float v8f;

// Fragment layout (ISA 7.12.2, 32-bit 16x4): lanes 0-15 hold K=k0,k0+1 in
// v0/v1; lanes 16-31 hold K=k0+2,k0+3.  M % 16 == 0 guaranteed by caller.

// ---------------------------------------------------------------------------
// Generic fp32 WMMA GEMM: C[M,Nc] = A[M,K=32] * B[K=32,Nc], strided views.
// One wave computes a 16 x (NT*16) strip; B fragment preloaded to registers.
// ---------------------------------------------------------------------------
template<int NT, bool CA>
__global__ __launch_bounds__(256) void gemm_wmma(
    const float* __restrict__ A, const float* __restrict__ B, float* __restrict__ C,
    int M, int Nc, int sAm, int sAk, int sBk, int sBn, int sCm, int sCn)
{
    const int lane  = threadIdx.x & 31;
    const int wave  = threadIdx.x >> 5;
    const int mTile = blockIdx.x * 8 + wave;
    if (mTile * 16 >= M) return;                  // wave-uniform exit
    const int l15   = lane & 15;
    const int khalf = (lane >> 4) << 1;           // 0 or 2
    const int row   = mTile * 16 + l15;
    const int aBase = row * sAm;

    // ---- preload this lane's B-fragment values (hoisted out of K-loop) ----
    float bf[NT][16];
    #pragma unroll
    for (int t = 0; t < NT; ++t) {
        const int col = t * 16 + l15;
        const bool cok = (col < Nc);
        const int  cc  = cok ? col : 0;            // clamp: load stays legal
        #pragma unroll
        for (int kk = 0; kk < 8; ++kk) {
            const int ka = kk * 4 + khalf;
            const float bx = B[ka * sBk + cc * sBn];
            const float by = B[(ka + 1) * sBk + cc * sBn];
            bf[t][2*kk]   = cok ? bx : 0.0f;       // v_cndmask, no exec branch
            bf[t][2*kk+1] = cok ? by : 0.0f;
        }
    }

    v8f acc[NT] = {};
    #pragma unroll
    for (int kk = 0; kk < 8; ++kk) {
        const int ka = kk * 4 + khalf;
        v2f a;
        if (CA) {                                  // contiguous K: one b64 load
            const float2 av = *(const float2*)(A + aBase + ka);
            a.x = av.x; a.y = av.y;
        } else {
            a.x = A[aBase + ka * sAk];
            a.y = A[aBase + (ka + 1) * sAk];
        }
        #pragma unroll
        for (int t = 0; t < NT; ++t) {
            v2f b; b.x = bf[t][2*kk]; b.y = bf[t][2*kk+1];
            acc[t] = __builtin_amdgcn_wmma_f32_16x16x4_f32(
                         false, a, false, b, (short)0, acc[t], false, false);
        }
    }
    const int rbase = mTile * 16 + ((lane >> 4) << 3);   // +8 for hi half
    #pragma unroll
    for (int t = 0; t < NT; ++t) {
        const int col = t * 16 + l15;
        if (col < Nc) {
            #pragma unroll
            for (int r = 0; r < 8; ++r)
                C[(rbase + r) * sCm + col * sCn] = acc[t][r];
        }
    }
}

// ---------------------------------------------------------------------------
// Dual-B fp32 WMMA GEMM: C1 = A@B1, C2 = A@B2.  Fixed K=32, Nc=32, all
// row-major contiguous.  A (the 128 MB edge-feature stream) is read ONCE.
// ---------------------------------------------------------------------------
__global__ __launch_bounds__(256) void gemm_wmma_x2(
    const float* __restrict__ A,
    const float* __restrict__ B1, const float* __restrict__ B2,
    float* __restrict__ C1, float* __restrict__ C2, int M)
{
    const int lane  = threadIdx.x & 31;
    const int wave  = threadIdx.x >> 5;
    const int mTile = blockIdx.x * 8 + wave;
    if (mTile * 16 >= M) return;
    const int l15   = lane & 15;
    const int khalf = (lane >> 4) << 1;
    const int row   = mTile * 16 + l15;
    const int aBase = row * 32;

    float bf1[2][16], bf2[2][16];
    #pragma unroll
    for (int t = 0; t < 2; ++t) {
        const int col = t * 16 + l15;
        #pragma unroll
        for (int kk = 0; kk < 8; ++kk) {
            const int ka = kk * 4 + khalf;
            bf1[t][2*kk]   = B1[ka * 32 + col];
            bf1[t][2*kk+1] = B1[(ka + 1) * 32 + col];
            bf2[t][2*kk]   = B2[ka * 32 + col];
            bf2[t][2*kk+1] = B2[(ka + 1) * 32 + col];
        }
    }

    v8f acc[4] = {};
    #pragma unroll
    for (int kk = 0; kk < 8; ++kk) {
        const int ka = kk * 4 + khalf;
        const float2 av = *(const float2*)(A + aBase + ka);
        v2f a; a.x = av.x; a.y = av.y;
        #pragma unroll
        for (int t = 0; t < 2; ++t) {
            v2f b; b.x = bf1[t][2*kk]; b.y = bf1[t][2*kk+1];
            acc[t] = __builtin_amdgcn_wmma_f32_16x16x4_f32(
                         false, a, false, b, (short)0, acc[t], false, false);
            v2f c; c.x = bf2[t][2*kk]; c.y = bf2[t][2*kk+1];
            acc[2+t] = __builtin_amdgcn_wmma_f32_16x16x4_f32(
                         false, a, false, c, (short)0, acc[2+t], false, false);
        }
    }
    const int rbase = mTile * 16 + ((lane >> 4) << 3);
    #pragma unroll
    for (int t = 0; t < 2; ++t) {
        const int col = t * 16 + l15;
        #pragma unroll
        for (int r = 0; r < 8; ++r) {
            C1[(rbase + r) * 32 + col] = acc[t][r];
            C2[(rbase + r) * 32 + col] = acc[2+t][r];
        }
    }
}

// ---- order-preserving float<->uint map for atomic max --------------------
__device__ __forceinline__ unsigned int fmap(float f) {
    unsigned int b = __float_as_uint(f);
    return (b & 0x80000000u) ? ~b : (b | 0x80000000u);
}
__device__ __forceinline__ float funmap(unsigned int u) {
    return (u & 0x80000000u) ? __uint_as_float(u & 0x7fffffffu)
                             : __uint_as_float(~u);
}

// ---- pass 1: logits + per-(node,head) max --------------------------------
__global__ __launch_bounds__(256) void edge_logits(
    const int* __restrict__ src, const int* __restrict__ dst,
    const float* __restrict__ Qn, const float* __restrict__ Kn,
    const float* __restrict__ EK, float* __restrict__ LG,
    unsigned int* __restrict__ Mbuf, int E)
{
    const int e = blockIdx.x * 256 + threadIdx.x;
    if (e >= E) return;
    const int s = src[e], d = dst[e];
    const float4* q4 = (const float4*)(Qn + d * 32);
    const float4* k4 = (const float4*)(Kn + s * 32);
    const float4* e4 = (const float4*)(EK + e * 32);
    const float scale = 0.35355339059327379f;   // 1/sqrt(8)
    #pragma unroll
    for (int h = 0; h < 4; ++h) {
        float4 qa = q4[h*2], qb = q4[h*2+1];
        float4 ka = k4[h*2], kb = k4[h*2+1];
        float4 ea = e4[h*2], eb = e4[h*2+1];
        float lg = qa.x*(ka.x+ea.x) + qa.y*(ka.y+ea.y)
                 + qa.z*(ka.z+ea.z) + qa.w*(ka.w+ea.w)
                 + qb.x*(kb.x+eb.x) + qb.y*(kb.y+eb.y)
                 + qb.z*(kb.z+eb.z) + qb.w*(kb.w+eb.w);
        lg *= scale;
        LG[e * 4 + h] = lg;
        atomicMax(&Mbuf[d * 4 + h], fmap(lg));
    }
}

// ---- pass 2: exp + denominator -------------------------------------------
__global__ __launch_bounds__(256) void edge_expsum(
    const int* __restrict__ dst, float* __restrict__ LG,
    const unsigned int* __restrict__ Mbuf, float* __restrict__ Dbuf, int E)
{
    const int e = blockIdx.x * 256 + threadIdx.x;
    if (e >= E) return;
    const int d = dst[e];
    #pragma unroll
    for (int h = 0; h < 4; ++h) {
        float m  = funmap(Mbuf[d * 4 + h]);
        float ev = expf(LG[e * 4 + h] - m);
        LG[e * 4 + h] = ev;
        atomicAdd(&Dbuf[d * 4 + h], ev);
    }
}

// ---- pass 3: normalized weighted scatter of v0 / v1 ----------------------
template<bool HASF1, int C1O, int DSH>
__global__ __launch_bounds__(256) void edge_aggregate(
    const int* __restrict__ src, const int* __restrict__ dst,
    const float* __restrict__ coords,
    const float* __restrict__ Ebuf,  // exp values (pass-2 result)
    const float* __restrict__ Dbuf,
    const float* __restrict__ V0n, const float* __restrict__ EV,
    const float* __restrict__ A01n,
    const float* __restrict__ T0,    // [N,3,32]  (layer 2 only)
    const float* __restrict__ U,     // [N,C1O,3] (layer 2 only)
    float* __restrict__ Out0, float* __restrict__ Out1, int E)
{
    const int e = blockIdx.x * 256 + threadIdx.x;
    if (e >= E) return;
    const int s = src[e], d = dst[e];

    float rx = coords[d*3+0] - coords[s*3+0];
    float ry = coords[d*3+1] - coords[s*3+1];
    float rz = coords[d*3+2] - coords[s*3+2];
    float inv = 1.0f / sqrtf(rx*rx + ry*ry + rz*rz + 1e-8f);
    float y0 = rx*inv, y1 = ry*inv, y2 = rz*inv;

    float alpha[4];
    #pragma unroll
    for (int h = 0; h < 4; ++h)
        alpha[h] = Ebuf[e*4+h] / (Dbuf[d*4+h] + 1e-9f);

    // degree-0 output
    const float* vs = V0n + s * 32;
    const float* ev = EV  + e * 32;
    const float* t0 = T0  + s * 96;
    #pragma unroll
    for (int c = 0; c < 32; ++c) {
        float v = vs[c] + ev[c];
        if (HASF1) v += y0*t0[c] + y1*t0[32+c] + y2*t0[64+c];
        atomicAdd(&Out0[d*32 + c], alpha[c >> 3] * v);
    }

    // degree-1 output
    const float* a01 = A01n + s * C1O;
    const float* us  = U + s * C1O * 3;
    #pragma unroll
    for (int c = 0; c < C1O; ++c) {
        float a  = alpha[c >> DSH];
        float b  = a01[c];
        float w0 = b*y0, w1 = b*y1, w2 = b*y2;
        if (HASF1) { w0 += us[c*3+0]; w1 += us[c*3+1]; w2 += us[c*3+2]; }
        int o = (d * C1O + c) * 3;
        atomicAdd(&Out1[o+0], a*w0);
        atomicAdd(&Out1[o+1], a*w1);
        atomicAdd(&Out1[o+2], a*w2);
    }
}

// ---- finalizers ----------------------------------------------------------
__global__ __launch_bounds__(256) void fin_relu_add(
    const float* __restrict__ a, const float* __restrict__ b,
    float* __restrict__ o, int n)
{
    int i = blockIdx.x * 256 + threadIdx.x;
    if (i < n) { float v = a[i] + b[i]; o[i] = v > 0.0f ? v : 0.0f; }
}
__global__ __launch_bounds__(256) void fin_add(
    const float* __restrict__ a, const float* __restrict__ b,
    float* __restrict__ o, int n)
{
    int i = blockIdx.x * 256 + threadIdx.x;
    if (i < n) o[i] = a[i] + b[i];
}

// ---- output heads (small matmuls, one thread per node) -------------------
__global__ __launch_bounds__(256) void heads_kernel(
    const float* __restrict__ h0,
    const float* __restrict__ Wwo, const float* __restrict__ bwo,
    const float* __restrict__ Wwb,
    const float* __restrict__ Wwc1, const float* __restrict__ bwc1,
    const float* __restrict__ Wwc2,
    const float* __restrict__ Wab, const float* __restrict__ bab,
    const float* __restrict__ Wc,
    float* __restrict__ out, int N)
{
    const int n = blockIdx.x * 256 + threadIdx.x;
    if (n >= N) return;
    float x[32];
    const float4* h4 = (const float4*)(h0 + n * 32);
    #pragma unroll
    for (int i = 0; i < 8; ++i) {
        float4 t = h4[i];
        x[4*i] = t.x; x[4*i+1] = t.y; x[4*i+2] = t.z; x[4*i+3] = t.w;
    }
    const int OFF_WB = N*8,  OFF_WC = N*16;
    const int OFF_AM = N*31, OFF_CS = N*32;

    #pragma unroll
    for (int c = 0; c < 8; ++c) {
        float awo = bwo[c], awb = 0.0f;
        #pragma unroll
        for (int k = 0; k < 32; ++k) {
            awo += x[k] * Wwo[k*8+c];
            awb += x[k] * Wwb[k*8+c];
        }
        out[n*8 + c]          = tanhf(awo);
        out[OFF_WB + n*8 + c] = tanhf(awb);
    }
    float hid[32];
    #pragma unroll
    for (int j = 0; j < 32; ++j) {
        float a = bwc1[j];
        #pragma unroll
        for (int k = 0; k < 32; ++k) a += x[k] * Wwc1[k*32+j];
        hid[j] = a;
    }
    for (int t = 0; t < 15; ++t) {
        float a = 0.0f;
        #pragma unroll
        for (int j = 0; j < 32; ++j) a += hid[j] * Wwc2[j*15+t];
        out[OFF_WC + n*15 + t] = fmaxf(a, 0.0f);
    }
    { // ampl = relu(h0 @ Wab + bab)
        float a = bab[0];
        #pragma unroll
        for (int k = 0; k < 32; ++k) a += x[k] * Wab[k];
        out[OFF_AM + n] = fmaxf(a, 0.0f);
    }
    for (int t = 0; t < 15; ++t) {
        float a0 = 0.0f, a1 = 0.0f;
        #pragma unroll
        for (int k = 0; k < 32; ++k) {
            a0 += x[k] * Wc[(t*32+k)*2+0];
            a1 += x[k] * Wc[(t*32+k)*2+1];
        }
        int o = OFF_CS + (t * N + n) * 2;
        out[o] = a0; out[o+1] = a1;
    }
}

// ===========================================================================
extern "C" void kernel_launch(void* const* d_in, const int* in_sizes, int n_in,
                              void* d_out, int out_size, void* d_ws, size_t ws_size,
                              hipStream_t stream)
{
    (void)out_size; (void)ws_size;
    const float* f0     = (const float*)d_in[0];
    const float* coords = (const float*)d_in[1];
    const float* ef     = (const float*)d_in[2];
    const int N = in_sizes[0] / 32;
    const int E = in_sizes[2] / 32;
    const int* esrc = (const int*)d_in[n_in-2];
    const int* edst = (const int*)d_in[n_in-1];

    // layer-1 params
    const float* Wq1   = (const float*)d_in[3];
    const float* Wk1   = (const float*)d_in[4];
    const float* Wek1  = (const float*)d_in[5];
    const float* Wv01  = (const float*)d_in[6];
    const float* Wev1  = (const float*)d_in[7];
    const float* W011  = (const float*)d_in[8];
    const float* Wsf01 = (const float*)d_in[9];
    // layer-2 params
    const float* Wq2   = (const float*)d_in[10];
    const float* Wk2   = (const float*)d_in[11];
    const float* Wek2  = (const float*)d_in[12];
    const float* Wv02  = (const float*)d_in[13];
    const float* Wev2  = (const float*)d_in[14];
    const float* W012  = (const float*)d_in[15];
    const float* Wsf02 = (const float*)d_in[16];
    const float* W10   = (const float*)d_in[17];
    const float* W11   = (const float*)d_in[18];
    const float* Wsf1  = (const float*)d_in[19];
    // heads
    const float* Wwo  = (const float*)d_in[20];
    const float* bwo  = (const float*)d_in[21];
    const float* Wwb  = (const float*)d_in[22];
    const float* Wwc1 = (const float*)d_in[23];
    const float* bwc1 = (const float*)d_in[24];
    const float* Wwc2 = (const float*)d_in[25];
    const float* Wab  = (const float*)d_in[26];
    const float* bab  = (const float*)d_in[27];
    const float* Wc   = (const float*)d_in[28];

    // ---- workspace carving (~350 MB) ----
    float* w = (float*)d_ws;
    float* EK   = w; w += (size_t)E*32;
    float* EVb  = w; w += (size_t)E*32;
    float* LG   = w; w += (size_t)E*4;
    float* Qn   = w; w += (size_t)N*32;
    float* Kn   = w; w += (size_t)N*32;
    float* V0n  = w; w += (size_t)N*32;
    float* Sn   = w; w += (size_t)N*32;
    float* A01n = w; w += (size_t)N*32;
    unsigned int* Mbuf = (unsigned int*)w; w += (size_t)N*4;
    float* Dbuf = w; w += (size_t)N*4;
    float* Out0 = w; w += (size_t)N*32;
    float* Out1 = w; w += (size_t)N*96;   // layer1 h1 [N,32,3]; layer2 [N,8,3]
    float* H0a  = w; w += (size_t)N*32;
    float* T0   = w; w += (size_t)N*96;   // [N,3,32]
    float* Ub   = w; w += (size_t)N*24;   // [N,8,3]
    float* SUb  = w; w += (size_t)N*24;   // [N,8,3]

    float* out = (float*)d_out;
    float* oH0 = out + (size_t)N*62;
    float* oH1 = out + (size_t)N*94;

    auto gemm = [&](const float* A, const float* B, float* C, int M, int Nc,
                    int sAm, int sAk, int sBk, int sBn, int sCm, int sCn) {
        dim3 grid((unsigned)(((M + 15)/16 + 7)/8));
        dim3 blk(256);
        const bool ca = (sAk == 1);
        if (Nc > 16) {
            if (ca) gemm_wmma<2,true ><<<grid, blk, 0, stream>>>(A,B,C,M,Nc,sAm,sAk,sBk,sBn,sCm,sCn);
            else    gemm_wmma<2,false><<<grid, blk, 0, stream>>>(A,B,C,M,Nc,sAm,sAk,sBk,sBn,sCm,sCn);
        } else {
            if (ca) gemm_wmma<1,true ><<<grid, blk, 0, stream>>>(A,B,C,M,Nc,sAm,sAk,sBk,sBn,sCm,sCn);
            else    gemm_wmma<1,false><<<grid, blk, 0, stream>>>(A,B,C,M,Nc,sAm,sAk,sBk,sBn,sCm,sCn);
        }
    };
    const int EB = (E + 255) / 256;
    const dim3 EG((unsigned)((E/16 + 7)/8));

    // ======================= layer 1 =======================
    gemm(f0, Wq1,   Qn,   N, 32, 32,1, 32,1, 32,1);
    gemm(f0, Wk1,   Kn,   N, 32, 32,1, 32,1, 32,1);
    gemm(f0, Wv01,  V0n,  N, 32, 32,1, 32,1, 32,1);
    gemm(f0, W011,  A01n, N, 32, 32,1, 32,1, 32,1);
    gemm(f0, Wsf01, Sn,   N, 32, 32,1, 32,1, 32,1);
    gemm_wmma_x2<<<EG, 256, 0, stream>>>(ef, Wek1, Wev1, EK, EVb, E);

    hipMemsetAsync(Mbuf, 0, (size_t)N*4*sizeof(unsigned int), stream);
    hipMemsetAsync(Dbuf, 0, (size_t)N*4*sizeof(float), stream);
    hipMemsetAsync(Out0, 0, (size_t)N*32*sizeof(float), stream);
    hipMemsetAsync(Out1, 0, (size_t)N*96*sizeof(float), stream);

    edge_logits<<<EB, 256, 0, stream>>>(esrc, edst, Qn, Kn, EK, LG, Mbuf, E);
    edge_expsum<<<EB, 256, 0, stream>>>(edst, LG, Mbuf, Dbuf, E);
    edge_aggregate<false,32,3><<<EB, 256, 0, stream>>>(
        esrc, edst, coords, LG, Dbuf, V0n, EVb, A01n, T0, Ub, Out0, Out1, E);
    fin_relu_add<<<(N*32+255)/256, 256, 0, stream>>>(Out0, Sn, H0a, N*32);
    // layer-1 h1 lives in Out1 ([N,32,3])

    // ======================= layer 2 =======================
    // degree-1 per-node precomputes from f1 = Out1 (strided views f1[:,:,i])
    for (int i = 0; i < 3; ++i) {
        gemm(Out1 + i, W10,  T0 + i*32, N, 32, 96,3, 32,1, 96,1);  // T0[n,i,d]
        gemm(Out1 + i, W11,  Ub  + i,   N,  8, 96,3,  8,1, 24,3);  // U[n,d,i]
        gemm(Out1 + i, Wsf1, SUb + i,   N,  8, 96,3,  8,1, 24,3);  // SelfU[n,d,i]
    }
    gemm(H0a, Wq2,   Qn,   N, 32, 32,1, 32,1, 32,1);
    gemm(H0a, Wk2,   Kn,   N, 32, 32,1, 32,1, 32,1);
    gemm(H0a, Wv02,  V0n,  N, 32, 32,1, 32,1, 32,1);
    gemm(H0a, W012,  A01n, N,  8, 32,1,  8,1,  8,1);
    gemm(H0a, Wsf02, Sn,   N, 32, 32,1, 32,1, 32,1);
    gemm_wmma_x2<<<EG, 256, 0, stream>>>(ef, Wek2, Wev2, EK, EVb, E);

    hipMemsetAsync(Mbuf, 0, (size_t)N*4*sizeof(unsigned int), stream);
    hipMemsetAsync(Dbuf, 0, (size_t)N*4*sizeof(float), stream);
    hipMemsetAsync(Out0, 0, (size_t)N*32*sizeof(float), stream);
    hipMemsetAsync(Out1, 0, (size_t)N*24*sizeof(float), stream);  // after T0/U/SelfU

    edge_logits<<<EB, 256, 0, stream>>>(esrc, edst, Qn, Kn, EK, LG, Mbuf, E);
    edge_expsum<<<EB, 256, 0, stream>>>(edst, LG, Mbuf, Dbuf, E);
    edge_aggregate<true,8,1><<<EB, 256, 0, stream>>>(
        esrc, edst, coords, LG, Dbuf, V0n, EVb, A01n, T0, Ub, Out0, Out1, E);
    fin_relu_add<<<(N*32+255)/256, 256, 0, stream>>>(Out0, Sn, oH0, N*32);
    fin_add     <<<(N*24+255)/256, 256, 0, stream>>>(Out1, SUb, oH1, N*24);

    // ======================= heads =======================
    heads_kernel<<<(N+255)/256, 256, 0, stream>>>(oH0, Wwo, bwo, Wwb,
                                                  Wwc1, bwc1, Wwc2,
                                                  Wab, bab, Wc, out, N);
}